// AttGraphModel_79680233276023
// MI455X (gfx1250) — compile-verified
//
#include <hip/hip_runtime.h>
#include <cstdint>
#include <cstddef>

#define NNODES 30000
#define NEDGES 480000
#define HEADS  4
#define NEG_SLOPE 0.2f

typedef __attribute__((ext_vector_type(2))) float v2f;
typedef __attribute__((ext_vector_type(8))) float v8f;

static inline int cdiv(long long a, long long b) { return (int)((a + b - 1) / b); }

// ---- monotonic float<->uint encoding for atomic max over signed floats ----
__device__ __forceinline__ unsigned encF(float v) {
  unsigned u = __float_as_uint(v);
  return (u & 0x80000000u) ? ~u : (u | 0x80000000u);
}
__device__ __forceinline__ float decF(unsigned k) {
  unsigned u = (k & 0x80000000u) ? (k & 0x7FFFFFFFu) : ~k;
  return __uint_as_float(u);
}
#define ENC_NEG_INF 0x007FFFFFu   // encF(-inf)

// =====================================================================
// fp32 GEMM via V_WMMA_F32_16X16X4_F32: Y(n x kout) = X(n x kin) @ W
// One wave per 16x16 C tile; K stepped by 4. n%16==0, kin%4==0, kout%16==0.
// =====================================================================
__global__ void gemm_wmma_f32(const float* __restrict__ X,
                              const float* __restrict__ W,
                              float* __restrict__ Y,
                              int n, int kin, int kout) {
  const int lane   = threadIdx.x & 31;
  const int wave   = threadIdx.x >> 5;
  const int tilesN = kout >> 4;
  const int tile   = blockIdx.x * (blockDim.x >> 5) + wave;
  if (tile >= (n >> 4) * tilesN) return;        // uniform per wave

  const int tm   = tile / tilesN;
  const int tn   = tile % tilesN;
  const int l15  = lane & 15;
  const int koff = (lane >> 4) << 1;            // lanes 16-31 carry K+2,K+3
  const int col  = tn * 16 + l15;               // B / C column for this lane
  const float* __restrict__ Xr = X + (size_t)(tm * 16 + l15) * kin;

  v8f acc = {};
  for (int k = 0; k < kin; k += 4) {
    v2f a, b;
    a.x = Xr[k + koff];
    a.y = Xr[k + koff + 1];
    b.x = W[(size_t)(k + koff)     * kout + col];
    b.y = W[(size_t)(k + koff + 1) * kout + col];
    acc = __builtin_amdgcn_wmma_f32_16x16x4_f32(
        /*neg_a=*/false, a, /*neg_b=*/false, b,
        /*c_mod=*/(short)0, acc, /*reuse_a=*/false, /*reuse_b=*/false);
  }
  const int rbase = tm * 16 + ((lane >> 4) << 3);   // upper half-wave = M+8
#pragma unroll
  for (int r = 0; r < 8; ++r)
    Y[(size_t)(rbase + r) * kout + col] = acc[r];
}

// ---------------------------------------------------------------------
__global__ void fill_f32_k(float* p, float v, int n) {
  int i = blockIdx.x * blockDim.x + threadIdx.x;
  if (i < n) p[i] = v;
}
__global__ void fill_u32_k(unsigned* p, unsigned v, int n) {
  int i = blockIdx.x * blockDim.x + threadIdx.x;
  if (i < n) p[i] = v;
}

// el[n,h] = sum_c f[n,h,c]*al[h,c]  (same for er)
__global__ void elr_k(const float* __restrict__ f,
                      const float* __restrict__ al,
                      const float* __restrict__ ar,
                      float* __restrict__ el, float* __restrict__ er,
                      int n, int cper) {
  int i = blockIdx.x * blockDim.x + threadIdx.x;
  if (i >= n * HEADS) return;
  int node = i / HEADS, h = i % HEADS;
  const float* fp  = f  + (size_t)node * HEADS * cper + h * cper;
  const float* alp = al + h * cper;
  const float* arp = ar + h * cper;
  float sl = 0.f, sr = 0.f;
  for (int c = 0; c < cper; ++c) { float v = fp[c]; sl += v * alp[c]; sr += v * arp[c]; }
  el[i] = sl; er[i] = sr;
}

__device__ __forceinline__ float leaky(float v) { return v > 0.f ? v : NEG_SLOPE * v; }

// pass 1: segment max of leaky_relu(el[src]+er[dst]) over dst
__global__ void edge_max_k(const int* __restrict__ src, const int* __restrict__ dst,
                           const float* __restrict__ el, const float* __restrict__ er,
                           unsigned* __restrict__ emax) {
  int i = blockIdx.x * blockDim.x + threadIdx.x;
  if (i >= NEDGES * HEADS) return;
  int e = i / HEADS, h = i % HEADS;
  float v = leaky(el[src[e] * HEADS + h] + er[dst[e] * HEADS + h]);
  atomicMax(&emax[dst[e] * HEADS + h], encF(v));
}

// pass 2: a = exp(e - max); store a; segment-sum into esum
__global__ void edge_exp_k(const int* __restrict__ src, const int* __restrict__ dst,
                           const float* __restrict__ el, const float* __restrict__ er,
                           const unsigned* __restrict__ emax,
                           float* __restrict__ esum, float* __restrict__ abuf) {
  int i = blockIdx.x * blockDim.x + threadIdx.x;
  if (i >= NEDGES * HEADS) return;
  int e = i / HEADS, h = i % HEADS;
  int d = dst[e] * HEADS + h;
  float v = leaky(el[src[e] * HEADS + h] + er[d - h + h]);   // er[dst,h]
  float a = expf(v - decF(emax[d]));
  abuf[i] = a;
  atomicAdd(&esum[d], a);
}

// pass 3: acc[dst, fi] += a * f[src, fi]   (one thread per edge-feature)
__global__ void edge_agg_k(const int* __restrict__ src, const int* __restrict__ dst,
                           const float* __restrict__ abuf, const float* __restrict__ f,
                           float* __restrict__ acc, long long total, int fdim) {
  long long i = (long long)blockIdx.x * blockDim.x + threadIdx.x;
  if (i >= total) return;
  int e  = (int)(i / fdim);
  int fi = (int)(i % fdim);
  int h  = fi / (fdim / HEADS);
  float alpha = abuf[(size_t)e * HEADS + h];
  atomicAdd(&acc[(size_t)dst[e] * fdim + fi], alpha * f[(size_t)src[e] * fdim + fi]);
}

// pass 4: out = relu(acc / esum)  (zero for nodes with no in-edges)
__global__ void finalize_k(float* __restrict__ acc, const float* __restrict__ esum,
                           int n, int fdim) {
  int i = blockIdx.x * blockDim.x + threadIdx.x;
  if (i >= n * fdim) return;
  int node = i / fdim;
  int h    = (i % fdim) / (fdim / HEADS);
  float z  = esum[node * HEADS + h];
  float v  = (z > 0.f) ? acc[i] / z : 0.f;
  acc[i]   = v > 0.f ? v : 0.f;
}

// global max pool over rows (values are >=0 after relu; pooled pre-filled encF(0))
__global__ void pool_max_k(const float* __restrict__ x, unsigned* __restrict__ pooled,
                           int n, int fdim, int rows_per_blk) {
  int col = blockIdx.x * blockDim.x + threadIdx.x;
  if (col >= fdim) return;
  int r0 = blockIdx.y * rows_per_blk;
  int r1 = min(r0 + rows_per_blk, n);
  float m = 0.f;
  for (int r = r0; r < r1; ++r) m = fmaxf(m, x[(size_t)r * fdim + col]);
  atomicMax(&pooled[col], encF(m));
}

// pooled(512) @ fc_w(512x8) + fc_b -> softmax -> out(8)
__global__ void head_k(const unsigned* __restrict__ pooled,
                       const float* __restrict__ fcw, const float* __restrict__ fcb,
                       float* __restrict__ out, int kdim) {
  if (threadIdx.x != 0 || blockIdx.x != 0) return;
  float logits[8];
  for (int j = 0; j < 8; ++j) {
    float s = fcb[j];
    for (int k = 0; k < kdim; ++k) s += decF(pooled[k]) * fcw[k * 8 + j];
    logits[j] = s;
  }
  float m = logits[0];
  for (int j = 1; j < 8; ++j) m = fmaxf(m, logits[j]);
  float z = 0.f, e[8];
  for (int j = 0; j < 8; ++j) { e[j] = expf(logits[j] - m); z += e[j]; }
  for (int j = 0; j < 8; ++j) out[j] = e[j] / z;
}

// ---------------------------------------------------------------------
static void run_gat_layer(hipStream_t s,
                          const float* x, const float* W, const float* al, const float* ar,
                          const int* src, const int* dst,
                          float* fbuf, float* el, float* er, unsigned* emax,
                          float* esum, float* abuf, float* accOut,
                          int kin, int kout) {
  const int n = NNODES;
  const int tiles = (n / 16) * (kout / 16);
  gemm_wmma_f32<<<cdiv(tiles, 4), 128, 0, s>>>(x, W, fbuf, n, kin, kout);
  elr_k<<<cdiv((long long)n * HEADS, 256), 256, 0, s>>>(fbuf, al, ar, el, er, n, kout / HEADS);
  fill_u32_k<<<cdiv((long long)n * HEADS, 256), 256, 0, s>>>(emax, ENC_NEG_INF, n * HEADS);
  fill_f32_k<<<cdiv((long long)n * HEADS, 256), 256, 0, s>>>(esum, 0.f, n * HEADS);
  fill_f32_k<<<cdiv((long long)n * kout, 256), 256, 0, s>>>(accOut, 0.f, n * kout);
  edge_max_k<<<cdiv((long long)NEDGES * HEADS, 256), 256, 0, s>>>(src, dst, el, er, emax);
  edge_exp_k<<<cdiv((long long)NEDGES * HEADS, 256), 256, 0, s>>>(src, dst, el, er, emax, esum, abuf);
  long long total = (long long)NEDGES * kout;
  edge_agg_k<<<cdiv(total, 256), 256, 0, s>>>(src, dst, abuf, fbuf, accOut, total, kout);
  finalize_k<<<cdiv((long long)n * kout, 256), 256, 0, s>>>(accOut, esum, n, kout);
}

extern "C" void kernel_launch(void* const* d_in, const int* in_sizes, int n_in,
                              void* d_out, int out_size, void* d_ws, size_t ws_size,
                              hipStream_t stream) {
  (void)in_sizes; (void)n_in; (void)out_size; (void)ws_size;
  const float* x   = (const float*)d_in[0];
  const int*   src = (const int*)  d_in[1];
  const int*   dst = (const int*)  d_in[2];
  const float* W0  = (const float*)d_in[3];
  const float* al0 = (const float*)d_in[4];
  const float* ar0 = (const float*)d_in[5];
  const float* W1  = (const float*)d_in[6];
  const float* al1 = (const float*)d_in[7];
  const float* ar1 = (const float*)d_in[8];
  const float* W2  = (const float*)d_in[9];
  const float* al2 = (const float*)d_in[10];
  const float* ar2 = (const float*)d_in[11];
  const float* fcw = (const float*)d_in[12];
  const float* fcb = (const float*)d_in[13];
  float* out = (float*)d_out;

  // workspace carve (floats)
  float* ws = (float*)d_ws;
  size_t off = 0;
  float*    fbuf   = ws + off; off += (size_t)NNODES * 512;   // projected features
  float*    acc0   = ws + off; off += (size_t)NNODES * 128;   // layer0 out
  float*    acc1   = ws + off; off += (size_t)NNODES * 128;   // layer1 out
  float*    acc2   = ws + off; off += (size_t)NNODES * 512;   // layer2 out
  float*    el     = ws + off; off += (size_t)NNODES * HEADS;
  float*    er     = ws + off; off += (size_t)NNODES * HEADS;
  unsigned* emax   = (unsigned*)(ws + off); off += (size_t)NNODES * HEADS;
  float*    esum   = ws + off; off += (size_t)NNODES * HEADS;
  float*    abuf   = ws + off; off += (size_t)NEDGES * HEADS;
  unsigned* pooled = (unsigned*)(ws + off); off += 512;

  run_gat_layer(stream, x,    W0, al0, ar0, src, dst, fbuf, el, er, emax, esum, abuf, acc0, 128, 128);
  run_gat_layer(stream, acc0, W1, al1, ar1, src, dst, fbuf, el, er, emax, esum, abuf, acc1, 128, 128);
  run_gat_layer(stream, acc1, W2, al2, ar2, src, dst, fbuf, el, er, emax, esum, abuf, acc2, 128, 512);

  fill_u32_k<<<cdiv(512, 256), 256, 0, stream>>>(pooled, 0x80000000u /*encF(0)*/, 512);
  const int rows_per_blk = 256;
  dim3 pg(cdiv(512, 256), cdiv(NNODES, rows_per_blk));
  pool_max_k<<<pg, 256, 0, stream>>>(acc2, pooled, NNODES, 512, rows_per_blk);
  head_k<<<1, 1, 0, stream>>>(pooled, fcw, fcb, out, 512);
}